// Noisenet_35442070126718
// MI455X (gfx1250) — compile-verified
//
#include <hip/hip_runtime.h>

#define BN_EPS 1e-5f

typedef __attribute__((ext_vector_type(2))) float v2f;
typedef __attribute__((ext_vector_type(4))) float v4f;
typedef __attribute__((ext_vector_type(8))) float v8f;

// layer k: y = W_k (COUT[k] x CIN[k]) * a
constexpr int CIN[5]  = {1, 2, 4, 4, 2};
constexpr int COUT[5] = {2, 4, 4, 2, 1};

struct Params {
  const float* x;
  const float* w[5];
  const float* g[5];
  const float* b[5];
  float* out;
  float* stats;    // 40 floats: layer k -> scale[0..3] at k*8, shift[0..3] at k*8+4
  float* partial;  // NB * NS floats per pass (reused)
  int n;           // pixels = 32*512*512
};

// ---------------------------------------------------------------------------
// Per-pixel chain up to layer K-1 (input activation of layer K), VALU path.
// ---------------------------------------------------------------------------
template<int K>
__device__ __forceinline__ void chain_valu(float xv, const float wl[5][16],
                                           const float st[5][8], float a[4]) {
  a[0] = xv; a[1] = 0.f; a[2] = 0.f; a[3] = 0.f;
#pragma unroll
  for (int k = 0; k < K; k++) {
    float y[4] = {0.f, 0.f, 0.f, 0.f};
#pragma unroll
    for (int o = 0; o < COUT[k]; o++)
#pragma unroll
      for (int c = 0; c < CIN[k]; c++)
        y[o] = fmaf(wl[k][o * CIN[k] + c], a[c], y[o]);
#pragma unroll
    for (int o = 0; o < 4; o++)
      a[o] = (o < COUT[k]) ? fmaxf(fmaf(y[o], st[k][o], st[k][4 + o]), 0.f) : 0.f;
  }
}

// ---------------------------------------------------------------------------
// Pass k (k = 0..4): reduce Sum(a_{k-1}) and Sum(a_{k-1} a_{k-1}^T) over all
// pixels (a_{-1} = x). b128 loads, 4 pixels per thread per iteration.
// Deterministic: wave butterfly -> LDS -> per-block partial, no atomics.
// ---------------------------------------------------------------------------
template<int K>
__global__ __launch_bounds__(256) void stats_kernel(Params p) {
  constexpr int CI = CIN[K];
  constexpr int NS = CI + (CI * (CI + 1)) / 2;

  float wl[5][16];
  float st[5][8];
#pragma unroll
  for (int k = 0; k < K; k++) {
#pragma unroll
    for (int j = 0; j < 16; j++)
      wl[k][j] = (j < COUT[k] * CIN[k]) ? p.w[k][j] : 0.f;
#pragma unroll
    for (int j = 0; j < 8; j++)
      st[k][j] = p.stats[k * 8 + j];
  }

  float acc[NS];
#pragma unroll
  for (int j = 0; j < NS; j++) acc[j] = 0.f;

  int tid = blockIdx.x * blockDim.x + threadIdx.x;
  int stride = (gridDim.x * blockDim.x) * 4;
  for (int i = tid * 4; i + 3 < p.n; i += stride) {
    v4f xv = *(const v4f*)(p.x + i);
#pragma unroll
    for (int e = 0; e < 4; e++) {
      float a[4];
      chain_valu<K>(xv[e], wl, st, a);
      int q = 0;
#pragma unroll
      for (int c = 0; c < CI; c++) acc[q++] += a[c];
#pragma unroll
      for (int c = 0; c < CI; c++)
#pragma unroll
        for (int c2 = c; c2 < CI; c2++) acc[q++] += a[c] * a[c2];
    }
  }

  // wave32 butterfly reduction
#pragma unroll
  for (int j = 0; j < NS; j++)
#pragma unroll
    for (int off = 16; off > 0; off >>= 1)
      acc[j] += __shfl_xor(acc[j], off, 32);

  __shared__ float red[8][NS];
  int wid = threadIdx.x >> 5;
  if ((threadIdx.x & 31) == 0)
#pragma unroll
    for (int j = 0; j < NS; j++) red[wid][j] = acc[j];
  __syncthreads();
  if (threadIdx.x == 0) {
#pragma unroll
    for (int j = 0; j < NS; j++) {
      float s = 0.f;
      for (int w8 = 0; w8 < 8; w8++) s += red[w8][j];
      p.partial[(size_t)blockIdx.x * NS + j] = s;
    }
  }
}

// ---------------------------------------------------------------------------
// Finalize pass k: fold BN stats into scale/shift using
// mean_o = W_o . sumA / N,  E[y^2]_o = W_o^T S W_o / N,  var = E[y^2]-mean^2.
// Lane j sums partial component j (deterministic fixed-order sums).
// ---------------------------------------------------------------------------
template<int K>
__global__ void finalize_kernel(Params p, int nb) {
  constexpr int CI = CIN[K];
  constexpr int CO = COUT[K];
  constexpr int NS = CI + (CI * (CI + 1)) / 2;

  __shared__ float s[NS];
  int t = threadIdx.x;
  if (t < NS) {
    float acc = 0.f;
    for (int b = 0; b < nb; b++) acc += p.partial[(size_t)b * NS + t];
    s[t] = acc;
  }
  __syncthreads();
  if (t != 0 || blockIdx.x != 0) return;

  float sumA[4];
  float S[4][4];
  for (int c = 0; c < CI; c++) sumA[c] = s[c];
  int q = CI;
  for (int c = 0; c < CI; c++)
    for (int c2 = c; c2 < CI; c2++) { S[c][c2] = s[q]; S[c2][c] = s[q]; q++; }

  float invN = 1.f / (float)p.n;
  for (int o = 0; o < CO; o++) {
    float m = 0.f;
    for (int c = 0; c < CI; c++) m += p.w[K][o * CI + c] * sumA[c];
    m *= invN;
    float e2 = 0.f;
    for (int c = 0; c < CI; c++)
      for (int c2 = 0; c2 < CI; c2++)
        e2 += p.w[K][o * CI + c] * p.w[K][o * CI + c2] * S[c][c2];
    e2 *= invN;
    float var = e2 - m * m;
    float sc = p.g[K][o] * rsqrtf(var + BN_EPS);
    p.stats[K * 8 + o] = sc;
    p.stats[K * 8 + 4 + o] = p.b[K][o] - m * sc;
  }
  for (int o = CO; o < 4; o++) {
    p.stats[K * 8 + o] = 0.f;
    p.stats[K * 8 + 4 + o] = 0.f;
  }
}

// ---------------------------------------------------------------------------
// Output pass: full 5-layer chain per pixel via V_WMMA_F32_16X16X4_F32.
// A (16x4, zero-padded weights): lanes 0-15 = rows M, VGPR0/1 = K0/K1;
//                                lanes 16-31 = rows M, VGPR0/1 = K2/K3.
// B (4x16, 16 pixels):           lanes 0-15 hold ch0/ch1 in VGPR0/1 (cols N),
//                                lanes 16-31 hold ch2/ch3.
// D (16x16): VGPR o, lanes 0-15 = y[o][pixel].
// Lane L owns pixels base+4L..base+4L+3 (b128 load + b128 NT store); each of
// the 4 vector elements runs two 16-pixel WMMA groups. All control flow is
// wave-uniform so EXEC is all-ones for every WMMA.
// ---------------------------------------------------------------------------
__global__ __launch_bounds__(256) void out_kernel(Params p) {
  int lane = threadIdx.x & 31;

  float st[5][8];
#pragma unroll
  for (int k = 0; k < 5; k++)
#pragma unroll
    for (int j = 0; j < 8; j++) st[k][j] = p.stats[k * 8 + j];

  // Per-layer A matrices (weight tiles), constant across the loop.
  v2f A[5];
  int m = lane & 15;
  int cb = (lane < 16) ? 0 : 2;
#pragma unroll
  for (int k = 0; k < 5; k++) {
    float a0 = 0.f, a1 = 0.f;
    if (m < COUT[k]) {
      if (cb + 0 < CIN[k]) a0 = p.w[k][m * CIN[k] + cb + 0];
      if (cb + 1 < CIN[k]) a1 = p.w[k][m * CIN[k] + cb + 1];
    }
    A[k].x = a0;
    A[k].y = a1;
  }

  int gw = (blockIdx.x * blockDim.x + threadIdx.x) >> 5;
  int nw = (gridDim.x * blockDim.x) >> 5;
  for (int base = gw * 128; base + 127 < p.n; base += nw * 128) {
    v4f xv = *(const v4f*)(p.x + base + 4 * lane);
    v4f ov;
#pragma unroll
    for (int e = 0; e < 4; e++) {
      float xe = xv[e];
      float xs = __shfl_xor(xe, 16, 32);  // group-1 pixels into lanes 0-15
      float outv[2];
#pragma unroll
      for (int gsel = 0; gsel < 2; gsel++) {
        float x0 = gsel ? xs : xe;
        v2f bmat;
        bmat.x = (lane < 16) ? x0 : 0.f;  // channel 0 = x, channels 1..3 = 0
        bmat.y = 0.f;
#pragma unroll
        for (int k = 0; k < 5; k++) {
          v8f c = {};
          v8f d = __builtin_amdgcn_wmma_f32_16x16x4_f32(
              false, A[k], false, bmat, (short)0, c, false, false);
          float a0 = fmaxf(fmaf(d[0], st[k][0], st[k][4]), 0.f);
          float a1 = (COUT[k] > 1) ? fmaxf(fmaf(d[1], st[k][1], st[k][5]), 0.f) : 0.f;
          float a2 = (COUT[k] > 2) ? fmaxf(fmaf(d[2], st[k][2], st[k][6]), 0.f) : 0.f;
          float a3 = (COUT[k] > 3) ? fmaxf(fmaf(d[3], st[k][3], st[k][7]), 0.f) : 0.f;
          float t2 = (COUT[k] > 2) ? __shfl_xor(a2, 16, 32) : 0.f;
          float t3 = (COUT[k] > 3) ? __shfl_xor(a3, 16, 32) : 0.f;
          bmat.x = (lane < 16) ? a0 : t2;  // repack into B-layout for next layer
          bmat.y = (lane < 16) ? a1 : t3;
        }
        outv[gsel] = bmat.x;  // final channel 0, valid in lanes 0-15
      }
      float o1 = __shfl_xor(outv[1], 16, 32);
      ov[e] = (lane < 16) ? outv[0] : o1;
    }
    __builtin_nontemporal_store(ov, (v4f*)(p.out + base + 4 * lane));
  }
}

extern "C" void kernel_launch(void* const* d_in, const int* in_sizes, int n_in,
                              void* d_out, int out_size, void* d_ws, size_t ws_size,
                              hipStream_t stream) {
  (void)n_in; (void)out_size; (void)ws_size;
  Params p;
  p.x = (const float*)d_in[0];
  for (int i = 0; i < 5; i++) {
    p.w[i] = (const float*)d_in[1 + 3 * i];
    p.g[i] = (const float*)d_in[2 + 3 * i];
    p.b[i] = (const float*)d_in[3 + 3 * i];
  }
  p.out = (float*)d_out;
  float* ws = (float*)d_ws;
  p.stats = ws;
  p.partial = ws + 64;   // NB * NS_max * 4B = 1024*14*4 = 56 KB of scratch
  p.n = in_sizes[0];

  const int NB = 1024, BS = 256;
  stats_kernel<0><<<NB, BS, 0, stream>>>(p);
  finalize_kernel<0><<<1, 32, 0, stream>>>(p, NB);
  stats_kernel<1><<<NB, BS, 0, stream>>>(p);
  finalize_kernel<1><<<1, 32, 0, stream>>>(p, NB);
  stats_kernel<2><<<NB, BS, 0, stream>>>(p);
  finalize_kernel<2><<<1, 32, 0, stream>>>(p, NB);
  stats_kernel<3><<<NB, BS, 0, stream>>>(p);
  finalize_kernel<3><<<1, 32, 0, stream>>>(p, NB);
  stats_kernel<4><<<NB, BS, 0, stream>>>(p);
  finalize_kernel<4><<<1, 32, 0, stream>>>(p, NB);
  out_kernel<<<2048, BS, 0, stream>>>(p);
}